// QuantizedMemorySlots4bit_75728863363366
// MI455X (gfx1250) — compile-verified
//
#include <hip/hip_runtime.h>

// Row-wise 4-bit symmetric quantize -> dequantize (pack/unpack is identity on q).
// Inputs: memory_mean, memory_logvar, each (65536, 512) f32.
// Output: [mean_deq ; lv_deq] concatenated, f32.
//
// Roofline: 256 MB read + 256 MB write = 512 MB @ 23.3 TB/s ~= 22 us. Pure stream.
// Data movement uses the CDNA5 Tensor Data Mover: one TENSOR_LOAD_TO_LDS per
// 16 KB tile (8 rows), double-buffered in LDS, synchronized via TENSORcnt.

typedef float        v4f  __attribute__((ext_vector_type(4)));
typedef unsigned int u32x4 __attribute__((ext_vector_type(4)));
typedef int          i32x4 __attribute__((ext_vector_type(4)));
typedef int          i32x8 __attribute__((ext_vector_type(8)));

#define CODE       512          // floats per row
#define TILE_ROWS  8            // rows per tile == waves per block
#define TILE_ELEMS (TILE_ROWS * CODE)   // 4096 floats = 16 KB
#define QMAX_INV   (1.0f / 7.0f)

#if __has_builtin(__builtin_amdgcn_tensor_load_to_lds)
#define USE_TDM 1
#else
#define USE_TDM 0
#endif

#if USE_TDM
// Issue a TDM load of one 512x8 f32 tile (tile == its own tensor, so no OOB
// paths trigger). Descriptor bit-packing per CDNA5 ISA ch.8 (D# groups 0/1).
__device__ __forceinline__ void tdm_load_tile(const float* gsrc, unsigned lds_byte_off)
{
    unsigned long long ga = (unsigned long long)gsrc;
    u32x4 g0;
    g0[0] = 1u;                                   // count=1, user descriptor
    g0[1] = lds_byte_off;                         // lds_addr (bytes)
    g0[2] = (unsigned)(ga & 0xFFFFFFFFu);         // global_addr[31:0]
    g0[3] = (unsigned)((ga >> 32) & 0x01FFFFFFu)  // global_addr[56:32]
          | (2u << 30);                           // type=2 ("image")
    i32x8 g1;
    g1[0] = 0x00020000;        // workgroup_mask=0, data_size=2 (4 B), no flags
    g1[1] = (CODE << 16);      // atomic_barrier=0 ; tensor_dim0[15:0]=512
    g1[2] = (TILE_ROWS << 16); // tensor_dim0[31:16]=0 ; tensor_dim1[15:0]=8
    g1[3] = (CODE << 16);      // tensor_dim1[31:16]=0 ; tile_dim0=512
    g1[4] = TILE_ROWS;         // tile_dim1=8 ; tile_dim2=0
    g1[5] = CODE;              // tensor_dim0_stride[31:0]=512 (element units)
    g1[6] = 0;                 // stride0[47:32]=0 ; tensor_dim1_stride[15:0]=0
    g1[7] = 0;                 // tensor_dim1_stride[47:16]=0
    i32x4 g2 = {0, 0, 0, 0};
    i32x4 g3 = {0, 0, 0, 0};
#if __clang_major__ >= 23
    i32x8 g4 = {0, 0, 0, 0, 0, 0, 0, 0};
    __builtin_amdgcn_tensor_load_to_lds(g0, g1, g2, g3, g4, 0);
#else
    __builtin_amdgcn_tensor_load_to_lds(g0, g1, g2, g3, 0);
#endif
}
#endif // USE_TDM

__global__ __launch_bounds__(256) void qdq_tdm_kernel(
    const float* __restrict__ mean,
    const float* __restrict__ logvar,
    float* __restrict__ out,
    int tiles_per_tensor)          // 65536/8 = 8192
{
    __shared__ __align__(16) float lbuf[2][TILE_ELEMS];   // 2 x 16 KB ping-pong

    const int lane   = threadIdx.x & 31;     // wave32 lane
    const int wave   = threadIdx.x >> 5;     // 8 waves, one row each
    const int ntiles = 2 * tiles_per_tensor;
    const int stride = gridDim.x;

#if USE_TDM
    // Prologue: wave 0 kicks off the DMA for this block's first tile.
    if (threadIdx.x < 32 && (int)blockIdx.x < ntiles) {
        const float* s0 = ((int)blockIdx.x < tiles_per_tensor)
            ? (mean + (long long)blockIdx.x * TILE_ELEMS)
            : (logvar + (long long)((int)blockIdx.x - tiles_per_tensor) * TILE_ELEMS);
        tdm_load_tile(s0, (unsigned)(unsigned long long)&lbuf[0][0]);
    }
#endif

    int parity = 0;
    for (int T = blockIdx.x; T < ntiles; T += stride) {
#if USE_TDM
        if (threadIdx.x < 32) {
            __builtin_amdgcn_s_wait_tensorcnt(0);   // lbuf[parity] is resident
        }
        __syncthreads();                            // release the reader waves
        if (threadIdx.x < 32) {
            const int Tn = T + stride;
            if (Tn < ntiles) {                      // prefetch next tile (DMA
                const float* sn = (Tn < tiles_per_tensor)   //  overlaps compute)
                    ? (mean + (long long)Tn * TILE_ELEMS)
                    : (logvar + (long long)(Tn - tiles_per_tensor) * TILE_ELEMS);
                tdm_load_tile(sn, (unsigned)(unsigned long long)&lbuf[parity ^ 1][0]);
            }
        }
#else
        // Fallback: cooperative b128 loads into LDS (no TDM builtin available).
        {
            const float* s = (T < tiles_per_tensor)
                ? (mean + (long long)T * TILE_ELEMS)
                : (logvar + (long long)(T - tiles_per_tensor) * TILE_ELEMS);
            for (int i = threadIdx.x; i < TILE_ELEMS / 4; i += 256) {
                *(v4f*)&lbuf[parity][i * 4] =
                    __builtin_nontemporal_load((const v4f*)(s + i * 4));
            }
            __syncthreads();
        }
#endif
        // ---- compute row `wave` of tile T out of LDS ----
        const float* __restrict__ srow = &lbuf[parity][wave * CODE];
        float* __restrict__ drow =
            out + (long long)T * TILE_ELEMS + (long long)wave * CODE;

        // ds_load_b128: chunk j covers row bytes [j*512, j*512+512), lane L
        // takes 16 B at offset L*16 (matches the coalesced global store below).
        v4f x[4];
#pragma unroll
        for (int j = 0; j < 4; ++j) {
            x[j] = *(const v4f*)(srow + j * 128 + lane * 4);
        }

        float m = 0.0f;
#pragma unroll
        for (int j = 0; j < 4; ++j) {
#pragma unroll
            for (int k = 0; k < 4; ++k) {
                m = fmaxf(m, __builtin_fabsf(x[j][k]));
            }
        }
        // Wave32 cross-lane max reduction (5 steps).
#pragma unroll
        for (int off = 16; off > 0; off >>= 1) {
            m = fmaxf(m, __shfl_xor(m, off, 32));
        }

        const float scale = fmaxf(m * QMAX_INV, 1e-8f);
        const float inv   = 1.0f / scale;   // one IEEE division per row

#pragma unroll
        for (int j = 0; j < 4; ++j) {
            v4f y;
#pragma unroll
            for (int k = 0; k < 4; ++k) {
                float q = __builtin_rintf(x[j][k] * inv);  // RNE == jnp.round
                q = fminf(fmaxf(q, -8.0f), 7.0f);
                y[k] = q * scale;
            }
            __builtin_nontemporal_store(y, (v4f*)(drow + j * 128 + lane * 4));
        }

        parity ^= 1;
#if !USE_TDM
        __syncthreads();   // fallback path: protect lbuf reuse
#endif
        // TDM path: no trailing barrier needed — the next iteration's barrier
        // (after s_wait_tensorcnt) already orders buffer reuse: DMA into
        // buf[p^1] is only issued after every wave has passed a barrier that
        // follows its last read of buf[p^1].
    }
}

extern "C" void kernel_launch(void* const* d_in, const int* in_sizes, int n_in,
                              void* d_out, int out_size, void* d_ws, size_t ws_size,
                              hipStream_t stream) {
    const float* mean   = (const float*)d_in[0];   // memory_mean
    const float* logvar = (const float*)d_in[1];   // memory_logvar
    float* out = (float*)d_out;

    const int rows_per_tensor  = in_sizes[0] / CODE;            // 65536
    const int tiles_per_tensor = rows_per_tensor / TILE_ROWS;   // 8192
    const int ntiles           = 2 * tiles_per_tensor;          // 16384

    int blocks = 2048;                 // 8 tiles per block -> deep DMA pipeline
    if (blocks > ntiles) blocks = ntiles;

    qdq_tdm_kernel<<<blocks, 256, 0, stream>>>(mean, logvar, out,
                                               tiles_per_tensor);
}